// SphericalNSS_70909910057171
// MI455X (gfx1250) — compile-verified
//
#include <hip/hip_runtime.h>
#include <math.h>

// Problem constants (match reference setup_inputs)
constexpr int B = 16;
constexpr int H = 1024;
constexpr int W = 2048;
constexpr int F = 64;
constexpr float EPS = 1e-5f;

// Pass-1 decomposition: per batch, 64 blocks, 256 threads (8 waves), each wave
// owns a contiguous 4096-float chunk -> 32 iterations of float4 per lane.
constexpr int P1_CHUNKS = 64;
constexpr int P1_CHUNK = (H * W) / P1_CHUNKS;   // 32768 floats per block
constexpr int P1_WAVE_ELEMS = P1_CHUNK / 8;     // 4096 floats per wave

typedef __attribute__((ext_vector_type(2))) float v2f;
typedef __attribute__((ext_vector_type(8))) float v8f;

// ---------------------------------------------------------------------------
// Pass 1: per-batch sum(x) and sum(x^2) via WMMA reduction.
// D = A x Ones + D  => every column n of D accumulates: sum_m D[m][n] = sum(A).
// C/D layout: vgpr i, lanes 0..15 hold (m=i, n=lane); lanes 16..31 hold m=i+8.
// So per-lane sum of 8 vgprs + shfl_xor(16) gives the full column total.
// ---------------------------------------------------------------------------
__global__ __launch_bounds__(256)
void nss_pass1(const float* __restrict__ y,
               float* __restrict__ p1s, float* __restrict__ p1q) {
    const int b = blockIdx.y;
    const int chunk = blockIdx.x;
    const int tid = threadIdx.x;
    const int wave = tid >> 5;
    const int lane = tid & 31;

    const size_t base = (size_t)b * (size_t)(H * W)
                      + (size_t)chunk * P1_CHUNK
                      + (size_t)wave * P1_WAVE_ELEMS;
    const float4* p = reinterpret_cast<const float4*>(y + base);

    v8f accS = {};
    v8f accQ = {};
    v2f ones;
    ones[0] = 1.0f; ones[1] = 1.0f;

#pragma unroll 4
    for (int i = 0; i < P1_WAVE_ELEMS / 128; ++i) {     // 32 iterations
        float4 d = p[i * 32 + lane];
        v2f a0; a0[0] = d.x; a0[1] = d.y;
        v2f a1; a1[0] = d.z; a1[1] = d.w;
        accS = __builtin_amdgcn_wmma_f32_16x16x4_f32(false, a0, false, ones,
                                                     (short)0, accS, false, false);
        accS = __builtin_amdgcn_wmma_f32_16x16x4_f32(false, a1, false, ones,
                                                     (short)0, accS, false, false);
        v2f q0; q0[0] = d.x * d.x; q0[1] = d.y * d.y;
        v2f q1; q1[0] = d.z * d.z; q1[1] = d.w * d.w;
        accQ = __builtin_amdgcn_wmma_f32_16x16x4_f32(false, q0, false, ones,
                                                     (short)0, accQ, false, false);
        accQ = __builtin_amdgcn_wmma_f32_16x16x4_f32(false, q1, false, ones,
                                                     (short)0, accQ, false, false);
    }

    float s = accS[0] + accS[1] + accS[2] + accS[3]
            + accS[4] + accS[5] + accS[6] + accS[7];
    float q = accQ[0] + accQ[1] + accQ[2] + accQ[3]
            + accQ[4] + accQ[5] + accQ[6] + accQ[7];
    // pair lanes L and L+16: full column total (identical across columns)
    s += __shfl_xor(s, 16);
    q += __shfl_xor(q, 16);

    __shared__ float ws_s[8];
    __shared__ float ws_q[8];
    if (lane == 0) { ws_s[wave] = s; ws_q[wave] = q; }
    __syncthreads();
    if (tid == 0) {
        float S = 0.0f, Q = 0.0f;
        for (int w = 0; w < 8; ++w) { S += ws_s[w]; Q += ws_q[w]; }
        p1s[b * P1_CHUNKS + chunk] = S;
        p1q[b * P1_CHUNKS + chunk] = Q;
    }
}

// ---------------------------------------------------------------------------
// Pass 2: per (batch, fixation) block. The block owning the FIRST occurrence
// of a row reconstructs the final row value per column ("last covering
// fixation wins") and dots it with y_pred; other blocks write zeros.
// ---------------------------------------------------------------------------
__global__ __launch_bounds__(256)
void nss_pass2(const float* __restrict__ y,
               const float* __restrict__ fix,
               const float* __restrict__ ker,
               const int* __restrict__ klens,
               int Kmax,
               float* __restrict__ p2wf, float* __restrict__ p2ff) {
    const int b = blockIdx.y;
    const int f = blockIdx.x;
    const int tid = threadIdx.x;

    __shared__ int xs_s[F];
    __shared__ int ys_s[F];
    __shared__ int rowxs[F];
    __shared__ int nrow_s;

    if (tid < F) {
        float fxv = fix[((size_t)b * F + tid) * 2 + 0];
        float fyv = fix[((size_t)b * F + tid) * 2 + 1];
        xs_s[tid] = (int)rintf(fxv * (float)(W - 1));
        ys_s[tid] = (int)rintf(fyv * (float)(H - 1));
    }
    __syncthreads();

    const int yrow = ys_s[f];
    bool first = true;
    for (int j = 0; j < f; ++j) {
        if (ys_s[j] == yrow) { first = false; break; }
    }
    if (!first) {   // uniform across block
        if (tid == 0) { p2wf[b * F + f] = 0.0f; p2ff[b * F + f] = 0.0f; }
        return;
    }

    const float* yrowptr = y + ((size_t)b * H + yrow) * (size_t)W;
    float wf = 0.0f, ff = 0.0f;

    if (yrow == 0 || yrow == H - 1) {
        // pole row: all ones
        for (int p = tid; p < W; p += 256) {
            wf += yrowptr[p];
            ff += 1.0f;
        }
    } else {
        if (tid == 0) {
            int nr = 0;
            for (int j = 0; j < F; ++j)
                if (ys_s[j] == yrow) rowxs[nr++] = xs_s[j];
            nrow_s = nr;
        }
        __syncthreads();
        const int kl = klens[yrow];
        const int half = kl >> 1;
        const int nr = nrow_s;
        const float* krow = ker + (size_t)yrow * (size_t)Kmax;
        for (int p = tid; p < W; p += 256) {
            float val = 0.0f;
            for (int j = nr - 1; j >= 0; --j) {          // newest fixation wins
                int k = (p - (rowxs[j] - half)) & (W - 1);  // circular wrap
                if (k < kl) { val = krow[k]; break; }
            }
            wf += val * yrowptr[p];
            ff += val;
        }
    }

    __shared__ float rs[256];
    __shared__ float rf[256];
    rs[tid] = wf;
    rf[tid] = ff;
    __syncthreads();
    for (int s = 128; s > 0; s >>= 1) {
        if (tid < s) { rs[tid] += rs[tid + s]; rf[tid] += rf[tid + s]; }
        __syncthreads();
    }
    if (tid == 0) {
        p2wf[b * F + f] = rs[0];
        p2ff[b * F + f] = rf[0];
    }
}

// ---------------------------------------------------------------------------
// Finalize: deterministic fixed-order combination + NSS formula.
// ---------------------------------------------------------------------------
__global__ __launch_bounds__(32)
void nss_finalize(const float* __restrict__ ws, float* __restrict__ out) {
    __shared__ float nss[B];
    const int t = threadIdx.x;
    if (t < B) {
        const float* p1s  = ws;
        const float* p1q  = ws + B * P1_CHUNKS;
        const float* p2wf = ws + 2 * B * P1_CHUNKS;
        const float* p2ff = ws + 2 * B * P1_CHUNKS + B * F;

        float S = 0.0f, Q = 0.0f;
        for (int i = 0; i < P1_CHUNKS; ++i) {
            S += p1s[t * P1_CHUNKS + i];
            Q += p1q[t * P1_CHUNKS + i];
        }
        float WF = 0.0f, FF = 0.0f;
        for (int i = 0; i < F; ++i) {
            WF += p2wf[t * F + i];
            FF += p2ff[t * F + i];
        }
        const float N = (float)(H * W);
        float mean = S / N;
        float var = (Q - S * S / N) / (N - 1.0f);
        var = var > 0.0f ? var : 0.0f;
        float sd = sqrtf(var);
        float denom = sd + (sd < EPS ? EPS : 0.0f);
        nss[t] = (WF - mean * FF) / denom / (float)F;
    }
    __syncthreads();
    if (t == 0) {
        float a = 0.0f;
        for (int i = 0; i < B; ++i) a += nss[i];
        out[0] = a / (float)B;
    }
}

// ---------------------------------------------------------------------------
extern "C" void kernel_launch(void* const* d_in, const int* in_sizes, int n_in,
                              void* d_out, int out_size, void* d_ws, size_t ws_size,
                              hipStream_t stream) {
    const float* y    = (const float*)d_in[0];   // [B,1,H,W] f32
    const float* fix  = (const float*)d_in[1];   // [B,F,2]   f32
    const float* ker  = (const float*)d_in[2];   // [H,Kmax]  f32
    const int*   klen = (const int*)d_in[3];     // [H]       i32

    const int Hn   = in_sizes[3];
    const int Kmax = in_sizes[2] / Hn;

    float* ws   = (float*)d_ws;
    float* p1s  = ws;                            // [B*64]
    float* p1q  = ws + B * P1_CHUNKS;            // [B*64]
    float* p2wf = ws + 2 * B * P1_CHUNKS;        // [B*F]
    float* p2ff = ws + 2 * B * P1_CHUNKS + B * F;// [B*F]
    (void)p1q; (void)p2wf; (void)p2ff; (void)n_in; (void)out_size; (void)ws_size;

    nss_pass1<<<dim3(P1_CHUNKS, B), 256, 0, stream>>>(y, p1s, p1s + B * P1_CHUNKS);
    nss_pass2<<<dim3(F, B), 256, 0, stream>>>(y, fix, ker, klen, Kmax,
                                              ws + 2 * B * P1_CHUNKS,
                                              ws + 2 * B * P1_CHUNKS + B * F);
    nss_finalize<<<1, 32, 0, stream>>>(ws, (float*)d_out);
}